// WaveCell_12678743457841
// MI455X (gfx1250) — compile-verified
//
#include <hip/hip_runtime.h>
#include <math.h>

// ---------------------------------------------------------------------------
// Nonlinear 2-D wave FDTD, 256x256 grid, 4 batches, 256 sequential steps.
// One persistent workgroup (1024 thr = 32 waves, wave32) per batch.
// Chosen over the 512-thr variant: its 256-x-replayed time-loop body
// (~20 KB) fits the 32 KB WGP I$, and 8 waves/SIMD hide LDS/L2 latency.
//   - y(t)  : LDS 256x256 f32 field (256 KB)
//   - y(t-1): 40 VGPRs/lane (rows r8=1,2,5,6,7) + 64 KB LDS slab (rows 3,4)
//             + pipelined top row in regs  -> LDS = 320 KB exactly (WGP max)
//   - deferred row commit merges prev[]/yn[] buffers
//   - coefficients packed float2{proj,b_bound}: one b64 load per cell
//   - sched_barrier(ALU|DS) per cell caps scheduler-induced reg pressure
//   - CDNA5 TDM zero-fills the 64 KB y2 slab (tensor_load_to_lds from a
//     zeroed workspace region), overlapping the y1 zero-fill
// ---------------------------------------------------------------------------

#define NXY   256
#define TST   256
#define H2INV_F ((float)(1.0 / (2.0 * 1.01 * 1.01)))   // 1/H^2, H = sqrt(2)*1.01
#define BETA_F  100.0f
#define ETA_F   0.5f

__device__ __forceinline__ float fast_rcp(float v) {
#if __has_builtin(__builtin_amdgcn_rcpf)
  return __builtin_amdgcn_rcpf(v);
#else
  return 1.0f / v;
#endif
}

__device__ __forceinline__ void sched_fence() {
#if __has_builtin(__builtin_amdgcn_sched_barrier)
  __builtin_amdgcn_sched_barrier(0x81);   // ALU + DS may cross; VMEM may not
#endif
}

__device__ constexpr int y2idx(int r8) {  // r8 in {1,2,5,6,7} -> 0..4
  return (r8 < 3) ? (r8 - 1) : (r8 - 3);
}

// -------------------- per-cell update --------------------
// DT=1: inv_dt2=1, half_inv_dt=0.5 ; c_lin = 1 - 0.1*proj ; NL_C=0.1, NL_B0=0.5

__device__ __forceinline__ float cell_update(float c0, float nv, float sv,
                                             float wv, float ev, float y2v,
                                             float2 pb) {
  float p  = pb.x;
  float bb = pb.y;
  float lap  = H2INV_F * (nv + sv + wv + ev - 4.0f * c0);
  float c0sq = c0 * c0;
  float b    = bb + p * (0.5f * fast_rcp(1.0f + c0sq));
  float clin = 1.0f - 0.1f * p;
  float cc   = clin + 0.1f * p * c0sq;
  float num  = 2.0f * c0 - (1.0f - 0.5f * b) * y2v + cc * cc * lap;
  return num * fast_rcp(1.0f + 0.5f * b);
}

// -------------------- TDM plumbing --------------------

typedef unsigned int v4u __attribute__((ext_vector_type(4)));
typedef int          v4i __attribute__((ext_vector_type(4)));
typedef int          v8i __attribute__((ext_vector_type(8)));

#if __has_builtin(__builtin_amdgcn_tensor_load_to_lds) && \
    __has_builtin(__builtin_amdgcn_s_wait_tensorcnt)
#define USE_TDM 1
#endif

// -------------------- persistent simulation --------------------
// Wave w owns rows [8w, 8w+8). Per step:
//   merged phase : commit y_{t-1}[top] (ynT regs), compute y_t[top]
//   B1
//   interior     : rows base+1..base+7 top-down; row r-1's new values are
//                  committed while processing row r (after its north re-read)
//   B2

__global__ __launch_bounds__(1024) void simulate_kernel(
    const float* __restrict__ x, const float2* __restrict__ coef,
    const float* __restrict__ zbuf, float* __restrict__ out) {
  __shared__ float y1[NXY * NXY];      // 256 KB field
  __shared__ float y2l[64 * NXY];      // 64 KB: y(t-2) rows r8=3,4 per wave

  const int batch = blockIdx.x;
  const int tid   = threadIdx.x;
  const int w     = tid >> 5;     // wave id 0..31
  const int lane  = tid & 31;
  const int base  = w * 8;

  for (int i = tid; i < NXY * NXY; i += 1024) y1[i] = 0.0f;

#ifdef USE_TDM
  // DMA zero-fill of the 64 KB y2 slab: 1-row tile of 16384 4-byte elements
  // sourced from a zeroed workspace region (D# per CDNA5 ISA).
  if (tid < 32) {
    unsigned long long ga = (unsigned long long)(const void*)zbuf;
    unsigned lds_off = (unsigned)(unsigned long long)(const void*)&y2l[0];
    v4u g0;
    g0[0] = 1u;                                              // count=1
    g0[1] = lds_off;                                         // lds_addr
    g0[2] = (unsigned)(ga & 0xFFFFFFFFull);                  // global_addr lo
    g0[3] = (unsigned)(((ga >> 32) & 0x01FFFFFFull) | 0x80000000u); // hi|type=2
    v8i g1;
    g1[0] = (int)(2u << 16);            // wg_mask=0, data_size=4B (code 2)
    g1[1] = (int)((16384u & 0xFFFFu) << 16);  // tensor_dim0=16384 (lo16)
    g1[2] = (int)((16384u >> 16) | (1u << 16)); // dim0 hi | tensor_dim1=1
    g1[3] = (int)(16384u << 16);        // tile_dim0 = 16384
    g1[4] = 1;                          // tile_dim1 = 1, tile_dim2 = 0
    g1[5] = 16384;                      // tensor_dim0_stride
    g1[6] = 0;
    g1[7] = 0;
    v4i gz4 = {0, 0, 0, 0};
    v8i gz8 = {0, 0, 0, 0, 0, 0, 0, 0};
    __builtin_amdgcn_tensor_load_to_lds(g0, g1, gz4, gz4, gz8, 0);
    __builtin_amdgcn_s_wait_tensorcnt(0);
  }
#else
  for (int i = tid; i < 64 * NXY; i += 1024) y2l[i] = 0.0f;
#endif
  __syncthreads();

  float y2r[40];                  // y(t-2) rows r8 = 1,2,5,6,7
#pragma unroll
  for (int i = 0; i < 40; ++i) y2r[i] = 0.0f;
  float ynT[8], y2top[8];         // pipelined top row: y_{t-1}, y_{t-2}
#pragma unroll
  for (int k = 0; k < 8; ++k) { ynT[k] = 0.0f; y2top[k] = 0.0f; }

  float acc0 = 0.0f, acc1 = 0.0f, acc2 = 0.0f;
  const bool probeOwner = (w == 28) && (lane == 0);  // row 226, cols 64/128/192
  const bool srcOwner   = (w == 3)  && (lane == 0);  // row 30,  col 128

  for (int t = 0; t < TST; ++t) {
    float xi = x[batch * TST + t];      // uniform scalar load (L2/K$)
    const int rT = base;

    // ---- merged phase: commit previous top row, compute new top row ----
#pragma unroll
    for (int k = 0; k < 8; ++k)
      y1[rT * NXY + k * 32 + lane] = ynT[k];           // commit y_{t-1}[top]
#pragma unroll
    for (int k = 0; k < 8; ++k) {
      int col = k * 32 + lane;
      int idx = rT * NXY + col;
      float2 pb = coef[idx];
      float c0 = ynT[k];                               // y_{t-1}[top]
      float nv = (w > 0) ? y1[idx - NXY] : 0.0f;       // band above's bottom
      float sv = y1[idx + NXY];                        // own row base+1 (old)
      float wv = (col > 0)       ? y1[idx - 1] : 0.0f; // committed row (old)
      float ev = (col < NXY - 1) ? y1[idx + 1] : 0.0f;
      float y = cell_update(c0, nv, sv, wv, ev, y2top[k], pb);
      y2top[k] = c0;
      ynT[k] = y;                                      // y_t[top], in regs
      sched_fence();
    }
    __syncthreads();   // B1: all top-row commits + cross-band reads done

    // ---- interior rows top-down, deferred row commit ----
    float cur[8], ynP[8];
#pragma unroll
    for (int r8 = 1; r8 <= 7; ++r8) {
      const int r = base + r8;
#pragma unroll
      for (int k = 0; k < 8; ++k) cur[k] = y1[r * NXY + k * 32 + lane];
#pragma unroll
      for (int k = 0; k < 8; ++k) {
        int col = k * 32 + lane;
        int idx = r * NXY + col;
        float2 pb = coef[idx];
        float nv = y1[idx - NXY];             // row r-1, still old
        if (r8 >= 2) y1[idx - NXY] = ynP[k];  // commit row r-1 new value
        float sv = (r8 == 7) ? ((w < 31) ? y1[idx + NXY] : 0.0f)
                             : y1[idx + NXY]; // row r+1 still old
        float wv = (col > 0)       ? y1[idx - 1] : 0.0f;
        float ev = (col < NXY - 1) ? y1[idx + 1] : 0.0f;
        float y2v;
        if (r8 == 3 || r8 == 4)
          y2v = y2l[(w * 2 + (r8 - 3)) * NXY + col];
        else
          y2v = y2r[y2idx(r8) * 8 + k];
        float y = cell_update(cur[k], nv, sv, wv, ev, y2v, pb);
        if (srcOwner && r8 == 6 && k == 4) y += xi;    // source (30,128)
        if (r8 == 3 || r8 == 4)
          y2l[(w * 2 + (r8 - 3)) * NXY + col] = cur[k];
        else
          y2r[y2idx(r8) * 8 + k] = cur[k];
        ynP[k] = y;
        sched_fence();
      }
      if (probeOwner && r8 == 2) {        // row 226
        acc0 += ynP[2] * ynP[2];          // col 64
        acc1 += ynP[4] * ynP[4];          // col 128
        acc2 += ynP[6] * ynP[6];          // col 192
      }
    }
#pragma unroll
    for (int k = 0; k < 8; ++k)           // commit bottom row (base+7)
      y1[(base + 7) * NXY + k * 32 + lane] = ynP[k];
    __syncthreads();   // B2: band settled before next step's merged phase
  }

  if (probeOwner) {
    float s = acc0 + acc1 + acc2;
    float inv = 1.0f / s;
    out[batch * 3 + 0] = acc0 * inv;
    out[batch * 3 + 1] = acc1 * inv;
    out[batch * 3 + 2] = acc2 * inv;
  }
}

// -------------------- coefficient precompute --------------------

__device__ __forceinline__ float pml1d(int i) {
  // b_vals = 3 * linspace(0,1,21)^4 mirrored at both edges (PML_N=20)
  float d = 0.0f;
  if (i <= 20)       d = (float)(20 - i) * 0.05f;
  else if (i >= 235) d = (float)(i - 235) * 0.05f;
  float d2 = d * d;
  return 3.0f * d2 * d2;
}

__global__ __launch_bounds__(256) void precompute_kernel(
    const float* __restrict__ rho, float2* __restrict__ coef,
    float* __restrict__ zbuf) {
  int idx = blockIdx.x * blockDim.x + threadIdx.x;   // 65536 cells
  int i = idx >> 8, j = idx & 255;
  float c = rho[idx];
  float n = (i > 0)   ? rho[idx - NXY] : 0.0f;
  float s = (i < 255) ? rho[idx + NXY] : 0.0f;
  float w = (j > 0)   ? rho[idx - 1]   : 0.0f;
  float e = (j < 255) ? rho[idx + 1]   : 0.0f;
  float lpf = 0.5f * c + 0.125f * (n + s + w + e);
  float tbe = tanhf(BETA_F * ETA_F);
  float p = (tbe + tanhf(BETA_F * (lpf - ETA_F))) /
            (tbe + tanhf(BETA_F * (1.0f - ETA_F)));
  float bx = pml1d(i), by = pml1d(j);
  float2 pb;
  pb.x = p;
  pb.y = sqrtf(bx * bx + by * by);
  coef[idx] = pb;
  if (idx < 64 * NXY) zbuf[idx] = 0.0f;   // TDM zero source (64 KB)
}

// -------------------- host launcher --------------------

extern "C" void kernel_launch(void* const* d_in, const int* in_sizes, int n_in,
                              void* d_out, int out_size, void* d_ws,
                              size_t ws_size, hipStream_t stream) {
  const float* x   = (const float*)d_in[0];   // (4, 256)
  const float* rho = (const float*)d_in[1];   // (256, 256)
  float2* coef = (float2*)d_ws;               // 512 KB packed {proj, b_bound}
  float*  zbuf = (float*)d_ws + 2 * NXY * NXY;  // 64 KB zero source for TDM
  float*  out  = (float*)d_out;               // (4, 3)

  hipLaunchKernelGGL(precompute_kernel, dim3(256), dim3(256), 0, stream,
                     rho, coef, zbuf);
  hipLaunchKernelGGL(simulate_kernel, dim3(4), dim3(1024), 0, stream,
                     x, coef, zbuf, out);
}